// GeoMix1_33440615367377
// MI455X (gfx1250) — compile-verified
//
#include <hip/hip_runtime.h>
#include <hip/hip_bf16.h>

#define DF 128                 // feature dim
#define KP_STRIDE 288          // floats per k-pair row in LDS (256 data + 32 pad)
typedef __attribute__((ext_vector_type(2))) float v2f;
typedef __attribute__((ext_vector_type(8))) float v8f;

// ---------------------------------------------------------------------------
// Degree / normalization kernels
// ---------------------------------------------------------------------------
__global__ void deg_init_kernel(float* __restrict__ deg, int n) {
    int i = blockIdx.x * blockDim.x + threadIdx.x;
    if (i < n) deg[i] = 1.0f;   // analytic self loop contributes 1
}

__global__ void deg_accum_kernel(const long long* __restrict__ ei,
                                 float* __restrict__ deg, int E) {
    int e = blockIdx.x * blockDim.x + threadIdx.x;
    if (e < E) {
        long long s = ei[e];
        long long d = ei[E + e];
        if (s != d) atomicAdd(&deg[(int)d], 1.0f);
    }
}

__global__ void dinv_kernel(const float* __restrict__ deg,
                            float* __restrict__ dinv, int n) {
    int i = blockIdx.x * blockDim.x + threadIdx.x;
    if (i < n) dinv[i] = rsqrtf(deg[i]);
}

// scale = gamma * rsqrt(var+eps);  shift = beta - mean*scale
__global__ void bn_coef_kernel(const float* __restrict__ g,
                               const float* __restrict__ b,
                               const float* __restrict__ m,
                               const float* __restrict__ v,
                               float* __restrict__ scale,
                               float* __restrict__ shift, int n) {
    int i = blockIdx.x * blockDim.x + threadIdx.x;
    if (i < n) {
        float s = g[i] * rsqrtf(v[i] + 1e-5f);
        scale[i] = s;
        shift[i] = b[i] - m[i] * s;
    }
}

// ---------------------------------------------------------------------------
// Fused dense GEMM using V_WMMA_F32_16X16X4_F32.
//   A-load (FUSE_BN):   a = relu(a*bnscale[k] + bnshift[k])   (BN+ReLU fusion)
//   Y [r,:] = A[r,:] @ W + bias
//   Y2[r,:] = dinv[r]^2 * Y[r,:]          (self-loop init of the scatter dst)
//
// Block = 256 threads (8 waves), 128 rows per block. W is staged in LDS with
// a (k,k+1) pair-interleaved, bank-padded layout so each B fragment is one
// conflict-free ds_load_b64 straight into the WMMA operand pair (no movs).
// FUSE_BN is a template parameter so each instantiation is branch-free.
// ---------------------------------------------------------------------------
template <bool FUSE_BN>
__global__ __launch_bounds__(256) void gemm_fused_wmma(
    const float* __restrict__ X, const float* __restrict__ W,
    const float* __restrict__ bias,
    const float* __restrict__ bnscale, const float* __restrict__ bnshift,
    const float* __restrict__ dinv,
    float* __restrict__ Y, float* __restrict__ Y2,
    int nrows)
{
    extern __shared__ float smem[];
    float* sW2    = smem;                       // 64 * KP_STRIDE floats
    float* sScale = smem + 64 * KP_STRIDE;      // DF floats
    float* sShift = sScale + DF;                // DF floats

    const int tid = threadIdx.x;

    // ---- stage W, swizzled: sW2[kp*KP_STRIDE + n*2 + (k&1)] = W[k][n] ------
    for (int w = tid; w < (DF / 2) * (DF / 4); w += 256) {   // 2048 items
        const int kp = w >> 5;                 // k-pair 0..63
        const int n4 = w & 31;                 // float4 column group 0..31
        const float4 r0 = *reinterpret_cast<const float4*>(W + (2 * kp)     * DF + 4 * n4);
        const float4 r1 = *reinterpret_cast<const float4*>(W + (2 * kp + 1) * DF + 4 * n4);
        float* b = sW2 + kp * KP_STRIDE + 8 * n4;
        *reinterpret_cast<float2*>(b + 0) = make_float2(r0.x, r1.x);
        *reinterpret_cast<float2*>(b + 2) = make_float2(r0.y, r1.y);
        *reinterpret_cast<float2*>(b + 4) = make_float2(r0.z, r1.z);
        *reinterpret_cast<float2*>(b + 6) = make_float2(r0.w, r1.w);
    }
    if (FUSE_BN && tid < DF) {                 // stage BN coefficients (1 KB)
        sScale[tid] = bnscale[tid];
        sShift[tid] = bnshift[tid];
    }
    __syncthreads();

    const int wave = tid >> 5;
    const int lane = tid & 31;
    const int row0 = blockIdx.x * 128 + wave * 16;

    // A fragment (16x4 f32): lanes 0-15 -> K={0,1}, lanes 16-31 -> K={2,3}
    const int am = lane & 15;
    const int ak = (lane >> 4) << 1;           // 0 or 2
    // B fragment base: half-wave k-pair select + column*2 (pair interleave)
    const int bn = lane & 15;
    const float* __restrict__ sB = sW2 + (lane >> 4) * KP_STRIDE + bn * 2;

    int arow = row0 + am;
    if (arow >= nrows) arow = nrows - 1;       // branchless clamp; EXEC full
    const float* __restrict__ arowp = X + (size_t)arow * DF;

    v8f acc[8];
    #pragma unroll
    for (int nt = 0; nt < 8; ++nt) {
        float bv = bias[nt * 16 + bn];
        v8f c{};
        #pragma unroll
        for (int v = 0; v < 8; ++v) c[v] = bv;
        acc[nt] = c;
    }

    // A-fragment loader; BN+ReLU path compiled in only when FUSE_BN
    auto loadA = [&](int k) -> v2f {
        const float2 av = *reinterpret_cast<const float2*>(arowp + k + ak);
        v2f a;
        if (FUSE_BN) {
            const float2 sc = *reinterpret_cast<const float2*>(sScale + k + ak);
            const float2 sh = *reinterpret_cast<const float2*>(sShift + k + ak);
            a[0] = fmaxf(av.x * sc.x + sh.x, 0.0f);
            a[1] = fmaxf(av.y * sc.y + sh.y, 0.0f);
        } else {
            a[0] = av.x; a[1] = av.y;
        }
        return a;
    };

    v2f a_cur = loadA(0);
    for (int k0 = 0; k0 < DF; k0 += 4) {
        const int kn = (k0 + 4 < DF) ? (k0 + 4) : k0;   // 1-deep pipeline
        v2f a_nxt = loadA(kn);
        const float* __restrict__ sBk = sB + (k0 >> 1) * KP_STRIDE;
        #pragma unroll
        for (int nt = 0; nt < 8; ++nt) {
            const float2 bp = *reinterpret_cast<const float2*>(sBk + nt * 32);
            v2f b; b[0] = bp.x; b[1] = bp.y;
            acc[nt] = __builtin_amdgcn_wmma_f32_16x16x4_f32(
                false, a_cur, false, b, (short)0, acc[nt], false, false);
        }
        a_cur = a_nxt;
    }

    // C/D layout: lanes 0-15 -> rows row0+0..7, lanes 16-31 -> rows row0+8..15
    const int cn = lane & 15;
    const int mb = (lane >> 4) << 3;           // 0 or 8
    #pragma unroll
    for (int v = 0; v < 8; ++v) {
        const int r = row0 + mb + v;
        if (r < nrows) {
            float* __restrict__ yr = Y + (size_t)r * DF + cn;
            const float dw = dinv[r];
            const float ws = dw * dw;
            float* __restrict__ y2 = Y2 + (size_t)r * DF + cn;
            #pragma unroll
            for (int nt = 0; nt < 8; ++nt) {
                const float val = acc[nt][v];
                yr[nt * 16] = val;
                y2[nt * 16] = ws * val;
            }
        }
    }
}

// ---------------------------------------------------------------------------
// Edge scatter: one wave per edge; edge/degree lookups are scalarized via
// readfirstlane (SMEM path). Each lane handles a float4: coalesced 512B row
// gather of H[src,:], then 4 contiguous f32 atomics into OUT[dst,:].
// ---------------------------------------------------------------------------
__global__ __launch_bounds__(256) void scatter_edges_kernel(
    const long long* __restrict__ ei, const float* __restrict__ dinv,
    const float* __restrict__ H, float* __restrict__ OUT, int E)
{
    const int e  = blockIdx.x * 8 + (threadIdx.x >> 5);
    const int f4 = threadIdx.x & 31;           // 32 float4 chunks = 128 floats
    if (e < E) {
        const int eu = __builtin_amdgcn_readfirstlane(e);   // wave-uniform
        const long long s = ei[eu];
        const long long d = ei[E + eu];
        if (s != d) {
            const float w = dinv[(int)s] * dinv[(int)d];
            const float4 hv =
                reinterpret_cast<const float4*>(H + (size_t)s * DF)[f4];
            float* __restrict__ o = OUT + (size_t)d * DF + f4 * 4;
            atomicAdd(o + 0, w * hv.x);
            atomicAdd(o + 1, w * hv.y);
            atomicAdd(o + 2, w * hv.z);
            atomicAdd(o + 3, w * hv.w);
        }
    }
}

// ---------------------------------------------------------------------------
// Host orchestration
// ---------------------------------------------------------------------------
extern "C" void kernel_launch(void* const* d_in, const int* in_sizes, int n_in,
                              void* d_out, int out_size, void* d_ws, size_t ws_size,
                              hipStream_t stream) {
    const float*     x   = (const float*)d_in[0];
    const long long* ei  = (const long long*)d_in[1];   // int64 [2,E] flat
    const float*     W1  = (const float*)d_in[2];
    const float*     b1  = (const float*)d_in[3];
    const float*     W2  = (const float*)d_in[4];
    const float*     b2  = (const float*)d_in[5];
    const float*     bng = (const float*)d_in[6];
    const float*     bnb = (const float*)d_in[7];
    const float*     bnm = (const float*)d_in[8];
    const float*     bnv = (const float*)d_in[9];
    float*           out = (float*)d_out;

    const int N  = in_sizes[0] / DF;           // 100000
    const int E  = in_sizes[1] / 2;            // 1600000
    const int ND = N * DF;

    // Workspace: deg[N] | dinv[N] | scale[DF] | shift[DF] | bufA[ND] | bufB[ND]
    float* deg   = (float*)d_ws;
    float* dinv  = deg + N;
    float* scale = dinv + N;
    float* shift = scale + DF;
    float* bufA  = shift + DF;
    float* bufB  = bufA + (size_t)ND;

    const int T = 256;
    dim3 blkN((N + T - 1) / T);
    dim3 blkE((E + T - 1) / T);
    dim3 blkGemm((N + 127) / 128);
    dim3 blkScat((E + 7) / 8);
    const size_t lds = (64 * KP_STRIDE + 2 * DF) * sizeof(float);  // 73 KB

    // degree + normalization + BN coefficients
    deg_init_kernel<<<blkN, T, 0, stream>>>(deg, N);
    deg_accum_kernel<<<blkE, T, 0, stream>>>(ei, deg, E);
    dinv_kernel<<<blkN, T, 0, stream>>>(deg, dinv, N);
    bn_coef_kernel<<<1, DF, 0, stream>>>(bng, bnb, bnm, bnv, scale, shift, DF);

    // layer 1: Linear (WMMA) -> bufA, self-loop init -> bufB, then scatter
    gemm_fused_wmma<false><<<blkGemm, T, lds, stream>>>(
        x, W1, b1, scale, shift, dinv, bufA, bufB, N);
    scatter_edges_kernel<<<blkScat, T, 0, stream>>>(ei, dinv, bufA, bufB, E);

    // layer 2: BN+ReLU fused into A-loads; Linear -> bufA, init -> out, scatter
    gemm_fused_wmma<true><<<blkGemm, T, lds, stream>>>(
        bufB, W2, b2, scale, shift, dinv, bufA, out, N);
    scatter_edges_kernel<<<blkScat, T, 0, stream>>>(ei, dinv, bufA, out, E);
}